// ASCADRecombine_77137612636756
// MI455X (gfx1250) — compile-verified
//
#include <hip/hip_runtime.h>
#include <hip/hip_bf16.h>

typedef __attribute__((ext_vector_type(2))) float v2f;
typedef __attribute__((ext_vector_type(8))) float v8f;

#define N_  256
#define WPB 8   // waves (rows) per block

// ---------------- GF(256) log/antilog tables, built at compile time ----------
struct GfTables {
    short alog[768];   // periodically extended: alog[i] = alog[i % 255]
    short log[256];
};

static constexpr GfTables make_tables() {
    GfTables t{};
    int x = 1;
    for (int i = 0; i < 255; ++i) {
        t.alog[i] = (short)x;
        t.log[x]  = (short)i;
        int v = x << 1;
        if (v & 256) v ^= 283;
        x = v ^ x;
    }
    t.log[0] = 0;
    for (int i = 255; i < 768; ++i) t.alog[i] = t.alog[i - 255];
    return t;
}

__constant__ GfTables G = make_tables();

// ---------------- wave32 helpers --------------------------------------------
__device__ __forceinline__ float wave_max(float v) {
#pragma unroll
    for (int o = 16; o > 0; o >>= 1) v = fmaxf(v, __shfl_xor(v, o, 32));
    return v;
}
__device__ __forceinline__ float wave_sum(float v) {
#pragma unroll
    for (int o = 16; o > 0; o >>= 1) v += __shfl_xor(v, o, 32);
    return v;
}

// +/-1.0f from the parity of popcount(a & b): sign bit only, no float converts
__device__ __forceinline__ float had_sign(int a, int b) {
    unsigned p = (unsigned)(__popc(a & b) & 1);
    return __int_as_float(0x3f800000u | (p << 31));
}

// bank-conflict-avoiding pad for the 512-entry extended conv buffer
__device__ __forceinline__ int cpad(int t) { return t + (t >> 4); }

// One WHT half-stage on a 16x16 row-major LDS tile:  BUF <- (H16 * BUF)^T.
// Applying it twice yields  BUF <- H16 * BUF * H16  (H16 symmetric), i.e. WHT_256.
// A-operand = constant +/-1 Hadamard chunks held in registers (hA).
__device__ __forceinline__ void wht_stage(float* buf, const v2f hA[4], int lane) {
    const int n     = lane & 15;
    const int khalf = (lane >> 4) << 1;     // lanes 0-15 -> K 0/1, lanes 16-31 -> K 2/3
    v8f acc = {};
#pragma unroll
    for (int kb = 0; kb < 4; ++kb) {
        const int k0 = 4 * kb + khalf;
        v2f b;
        b.x = buf[(k0 + 0) * 16 + n];
        b.y = buf[(k0 + 1) * 16 + n];
        acc = __builtin_amdgcn_wmma_f32_16x16x4_f32(false, hA[kb], false, b,
                                                    (short)0, acc, false, false);
    }
    __builtin_amdgcn_wave_barrier();
    const int rbase = (lane >> 4) * 8;      // C/D layout: vgpr i -> rows i and i+8
#pragma unroll
    for (int i = 0; i < 8; ++i)
        buf[n * 16 + (rbase + i)] = acc[i]; // transposed scatter (in-place safe: DS in-order)
    __builtin_amdgcn_wave_barrier();
}

// log_softmax over 256 values held as 8 per lane (float4 pair), then store.
__device__ __forceinline__ void softmax_store(float4 lo, float4 hi, float* o, int lane) {
    float mx = fmaxf(fmaxf(fmaxf(lo.x, lo.y), fmaxf(lo.z, lo.w)),
                     fmaxf(fmaxf(hi.x, hi.y), fmaxf(hi.z, hi.w)));
    mx = wave_max(mx);
    float s = __expf(lo.x - mx) + __expf(lo.y - mx) + __expf(lo.z - mx) + __expf(lo.w - mx)
            + __expf(hi.x - mx) + __expf(hi.y - mx) + __expf(hi.z - mx) + __expf(hi.w - mx);
    s = wave_sum(s);
    const float lse = mx + __logf(s);
    float4 r0 = make_float4(lo.x - lse, lo.y - lse, lo.z - lse, lo.w - lse);
    float4 r1 = make_float4(hi.x - lse, hi.y - lse, hi.z - lse, hi.w - lse);
    float4* o4 = (float4*)o;
    o4[lane]      = r0;
    o4[lane + 32] = r1;
}

// ---------------- main kernel: one wave handles one batch row ---------------
__global__ __launch_bounds__(WPB * 32)
void ascad_recombine_kernel(const float* __restrict__ alpha,
                            const float* __restrict__ beta,
                            const float* __restrict__ maskd,
                            float* __restrict__ out, int B) {
    __shared__ __align__(16) float sW1[WPB][N_];   // WHT(masked)
    __shared__ __align__(16) float sW2[WPB][N_];   // WHT(beta) -> conv
    __shared__ __align__(16) float sAr[WPB][N_];   // raw alpha row
    __shared__ __align__(16) float sCx[WPB][544];  // conv in alog order, extended+padded
    __shared__ __align__(16) float sAp[WPB][288];  // alpha in alog order, shifted window
    __shared__ __align__(16) float sPr[WPB][N_];   // prod_alog

    const int wave = (int)(threadIdx.x >> 5);
    const int lane = (int)(threadIdx.x & 31);
    const int row  = (int)blockIdx.x * WPB + wave;
    if (row >= B) return;                           // wave-uniform

    float* W1 = sW1[wave];
    float* W2 = sW2[wave];
    float* Ar = sAr[wave];
    float* Cx = sCx[wave];
    float* Ap = sAp[wave];
    float* Pr = sPr[wave];

    // Constant +/-1 Hadamard A-operand chunks, reused by all 6 WHT stages.
    v2f hA[4];
    {
        const int m = lane & 15;
        const int khalf = (lane >> 4) << 1;
#pragma unroll
        for (int kb = 0; kb < 4; ++kb) {
            const int k0 = 4 * kb + khalf;
            hA[kb].x = had_sign(m, k0 + 0);
            hA[kb].y = had_sign(m, k0 + 1);
        }
    }

    const size_t base  = (size_t)row * N_;
    const size_t total = (size_t)B * N_;
    const float4* a4 = (const float4*)(alpha + base);
    const float4* b4 = (const float4*)(beta  + base);
    const float4* m4 = (const float4*)(maskd + base);

    // --- load rows, emit the three trivial log_softmax outputs, stage to LDS ---
    float4 alo = a4[lane], ahi = a4[lane + 32];     // keep alpha in regs for S_alpha
    ((float4*)Ar)[lane] = alo; ((float4*)Ar)[lane + 32] = ahi;
    softmax_store(alo, ahi, out + base, lane);                     // log_alpha

    {
        float4 lo = b4[lane], hi = b4[lane + 32];
        ((float4*)W2)[lane] = lo; ((float4*)W2)[lane + 32] = hi;
        softmax_store(lo, hi, out + total + base, lane);           // log_beta
    }
    {
        float4 lo = m4[lane], hi = m4[lane + 32];
        ((float4*)W1)[lane] = lo; ((float4*)W1)[lane + 32] = hi;
        softmax_store(lo, hi, out + 2 * total + base, lane);       // log_masked
    }
    __builtin_amdgcn_wave_barrier();

    // --- XOR convolution: conv = WHT(WHT(masked) * WHT(beta)) / 256 (all WMMA) ---
    wht_stage(W1, hA, lane); wht_stage(W1, hA, lane);
    wht_stage(W2, hA, lane); wht_stage(W2, hA, lane);
#pragma unroll
    for (int i = 0; i < 8; ++i) {
        const int t = lane + 32 * i;
        W2[t] = W1[t] * W2[t] * (1.0f / 256.0f);
    }
    __builtin_amdgcn_wave_barrier();
    wht_stage(W2, hA, lane); wht_stage(W2, hA, lane);   // W2 = conv

    // --- gather into GF(256) antilog order with periodic extension ---
#pragma unroll
    for (int i = 0; i < 16; ++i) {
        const int t = lane + 32 * i;                 // t in [0,512)
        Cx[cpad(t)] = W2[G.alog[t]];                 // conv_alog[t % 255]
    }
#pragma unroll
    for (int i = 0; i < 9; ++i) {
        const int t = lane + 32 * i;                 // t in [0,288), alog index < 528 < 768
        Ap[t] = Ar[G.alog[t + 240]];                 // alpha_alog[(t - 15) mod 255]
    }
    __builtin_amdgcn_wave_barrier();

    // S_alpha = sum(alpha[1:])
    float sa = alo.x + alo.y + alo.z + alo.w + ahi.x + ahi.y + ahi.z + ahi.w;
    sa = wave_sum(sa) - __shfl(alo.x, 0, 32);

    // --- 255-pt circular correlation as one 16x16x256 f32 matmul (64 WMMAs) ---
    // prod_alog[16m+n] = sum_j conv_alog[(j+16m)%255] * alpha_alog[(j-n)%255]
    v8f acc = {};
    {
        const int n     = lane & 15;
        const int khalf = (lane >> 4) << 1;
#pragma unroll 8
        for (int jb = 0; jb < 64; ++jb) {
            const int j0 = 4 * jb + khalf;
            v2f a, b;
            a.x = Cx[cpad(j0 + 0 + 16 * n)];         // A[m=n][k] = c_ext[j + 16m]
            a.y = Cx[cpad(j0 + 1 + 16 * n)];
            const int t0 = j0 - n + 15;              // B[k][n] = a_ext[j - n], zero at j==255
            const float bx = Ap[t0], by = Ap[t0 + 1];
            b.x = (j0 + 0 < 255) ? bx : 0.0f;
            b.y = (j0 + 1 < 255) ? by : 0.0f;
            acc = __builtin_amdgcn_wmma_f32_16x16x4_f32(false, a, false, b,
                                                        (short)0, acc, false, false);
        }
        __builtin_amdgcn_wave_barrier();
        const int rbase = (lane >> 4) * 8;
#pragma unroll
        for (int i = 0; i < 8; ++i)
            Pr[(rbase + i) * 16 + n] = acc[i];       // prod_alog, natural order
        __builtin_amdgcn_wave_barrier();
    }

    // --- assemble res (scatter through LOG) and final log_softmax ---
    const float c0S = W2[0] * sa;                    // res[0] = conv[0] * sum(alpha[1:])
    float4 rlo, rhi;
    {
        const int z0 = lane * 4;
        rlo.x = (z0 == 0) ? c0S : Pr[G.log[z0 + 0]];
        rlo.y = Pr[G.log[z0 + 1]];
        rlo.z = Pr[G.log[z0 + 2]];
        rlo.w = Pr[G.log[z0 + 3]];
        const int z1 = 128 + lane * 4;
        rhi.x = Pr[G.log[z1 + 0]];
        rhi.y = Pr[G.log[z1 + 1]];
        rhi.z = Pr[G.log[z1 + 2]];
        rhi.w = Pr[G.log[z1 + 3]];
    }
    softmax_store(rlo, rhi, out + 3 * total + base, lane);         // log_target
}

// ---------------- launcher ---------------------------------------------------
extern "C" void kernel_launch(void* const* d_in, const int* in_sizes, int n_in,
                              void* d_out, int out_size, void* d_ws, size_t ws_size,
                              hipStream_t stream) {
    const float* alpha = (const float*)d_in[0];
    const float* beta  = (const float*)d_in[1];
    const float* maskd = (const float*)d_in[2];
    float* out = (float*)d_out;
    const int B = in_sizes[0] / N_;
    const int blocks = (B + WPB - 1) / WPB;
    ascad_recombine_kernel<<<blocks, WPB * 32, 0, stream>>>(alpha, beta, maskd, out, B);
    (void)n_in; (void)out_size; (void)d_ws; (void)ws_size;
}